// MyModel_87522843560619
// MI455X (gfx1250) — compile-verified
//
#include <hip/hip_runtime.h>

// ConvLSTM fused step kernel for gfx1250 (MI455X).
// Memory-bound (~200 MB mandatory traffic @ 23.3 TB/s vs ~5 GFLOP): fuse the
// input conv into the recurrent scan (no xg materialization), fp32 WMMA
// 16x16x4 for both convs as one GEMM with concatenated K = 128 + 16 = 144.
// Loads are unconditional (clamped addresses + cndmask zeroing) so the
// compiler can batch them ahead of the WMMA chain instead of exec-branching.

#if defined(__HIP_DEVICE_COMPILE__)
#if !__has_builtin(__builtin_amdgcn_wmma_f32_16x16x4_f32)
#error "missing __builtin_amdgcn_wmma_f32_16x16x4_f32 on this toolchain"
#endif
#endif

typedef __attribute__((ext_vector_type(2))) float v2f;
typedef __attribute__((ext_vector_type(8))) float v8f;

#define Bsz 8
#define Tsz 16
#define Csz 16
#define Fsz 2
#define Hsz 128
#define Wsz 128
#define HW  (Hsz * Wsz)
#define KX  128     // im2col K for x-conv: 4*2*16
#define KTOT 144    // + 4*2*2 for h-conv
#define NPAD 16     // 8 gate channels padded to 16

__device__ __forceinline__ float hsig(float z) {
    return fminf(fmaxf(0.2f * z + 0.5f, 0.0f), 1.0f);
}

__global__ __launch_bounds__(256) void convlstm_step(
    const float* __restrict__ x,   // (B,T,C,H,W)
    const float* __restrict__ Wx,  // (4,2,C,8)  HWIO
    const float* __restrict__ Wh,  // (4,2,F,8)  HWIO
    float* __restrict__ out,       // (B,T,F,H,W); slice t-1 is h_prev
    float* __restrict__ cws,       // (B,F,H,W) cell state, in-place
    int t)
{
    __shared__ float sB[KTOT * NPAD];     // padded [Wx ; Wh], zero cols 8..15
    __shared__ float sAcc[8 * 16 * 16];   // per-wave 16x16 accumulator tiles

    const int tid = threadIdx.x;

    // ---- stage B matrix (weights) into LDS, K ordered (c, kh, kw) ----
    for (int i = tid; i < KTOT * NPAD; i += 256) {
        int k = i >> 4, n = i & 15;
        float v = 0.0f;
        if (n < 8) {
            if (k < KX) {
                int c = k >> 3, kh = (k >> 1) & 3, kw = k & 1;
                v = Wx[(((kh * 2 + kw) * Csz + c) << 3) + n];
            } else {
                int hk = k - KX;
                int c2 = hk >> 3, kh = (hk >> 1) & 3, kw = hk & 1;
                v = Wh[(((kh * 2 + kw) * Fsz + c2) << 3) + n];
            }
        }
        sB[i] = v;
    }
    __syncthreads();

    const int lane = tid & 31;
    const int wave = tid >> 5;
    const int tile = blockIdx.x * 8 + wave;   // 8192 tiles: B * (H*W/16)
    const int b    = tile >> 10;
    const int rem  = tile & 1023;
    const int y    = rem >> 3;                // row
    const int x0   = (rem & 7) << 4;          // 16-pixel tile start in W

    const int m    = lane & 15;               // M index (pixel) / N index
    const int hiK  = (lane >> 4) << 1;        // lanes 16-31 hold K+2,K+3
    const int col  = x0 + m;
    const bool c1ok  = (col + 1) < Wsz;       // SAME: pad_right = 1
    const int  coff1 = c1ok ? 1 : 0;          // clamped second-column offset

    // time-reversed input slab
    const float* ximg  = x   + (size_t)((b * Tsz + (Tsz - 1 - t)) * Csz) * HW;
    const float* hprev = out + (size_t)((b * Tsz + (t - 1)) * Fsz) * HW;

    v8f acc = {0.f, 0.f, 0.f, 0.f, 0.f, 0.f, 0.f, 0.f};

    // ---- x-conv A fragments: 32 K-steps of 4, all loads issued up front ----
    v2f afr[32];
#pragma unroll
    for (int s = 0; s < 32; ++s) {
        const int k0 = s << 2;
        const int kb = k0 + hiK;
        const int c  = k0 >> 3;
        const int kh = (kb >> 1) & 3;
        const int row = y + kh - 1;           // SAME: pad_top = 1, range y-1..y+2
        const bool rok = (row >= 0) && (row < Hsz);
        const int rowc = min(max(row, 0), Hsz - 1);   // clamped, always in-image
        const float* rp = ximg + (size_t)c * HW + rowc * Wsz + col;
        const float v0 = rp[0];                       // unconditional loads
        const float v1 = rp[coff1];
        afr[s].x = rok ? v0 : 0.0f;                   // kw = 0
        afr[s].y = (rok && c1ok) ? v1 : 0.0f;         // kw = 1
    }

    // ---- x-conv WMMA chain ----
#pragma unroll
    for (int s = 0; s < 32; ++s) {
        const int kb = (s << 2) + hiK;
        v2f bb;
        bb.x = sB[kb * NPAD + m];
        bb.y = sB[(kb + 1) * NPAD + m];
        acc = __builtin_amdgcn_wmma_f32_16x16x4_f32(
            false, afr[s], false, bb, (short)0, acc, false, false);
    }

    // ---- h-conv: 4 K-steps of 4 (skip at t==0, h0 == 0; uniform branch) ----
    if (t > 0) {
        v2f hfr[4];
#pragma unroll
        for (int s = 0; s < 4; ++s) {
            const int kb = (s << 2) + hiK;     // hk in [0,16)
            const int c2 = (s << 2) >> 3;
            const int kh = (kb >> 1) & 3;
            const int row = y + kh - 1;
            const bool rok = (row >= 0) && (row < Hsz);
            const int rowc = min(max(row, 0), Hsz - 1);
            const float* rp = hprev + (size_t)c2 * HW + rowc * Wsz + col;
            const float v0 = rp[0];
            const float v1 = rp[coff1];
            hfr[s].x = rok ? v0 : 0.0f;
            hfr[s].y = (rok && c1ok) ? v1 : 0.0f;
        }
#pragma unroll
        for (int s = 0; s < 4; ++s) {
            const int kb = KX + (s << 2) + hiK;
            v2f bb;
            bb.x = sB[kb * NPAD + m];
            bb.y = sB[(kb + 1) * NPAD + m];
            acc = __builtin_amdgcn_wmma_f32_16x16x4_f32(
                false, hfr[s], false, bb, (short)0, acc, false, false);
        }
    }

    // ---- spill accumulator tile so each lane can gather one pixel's gates ----
    // D layout: VGPR r, lanes 0-15 -> [M=r][N=lane], lanes 16-31 -> [M=r+8][N=lane-16]
    float* sw = sAcc + wave * 256;
    const int mbase = (lane < 16) ? 0 : 8;
#pragma unroll
    for (int r = 0; r < 8; ++r)
        sw[(mbase + r) * 16 + m] = acc[r];
    __syncthreads();

    // lane -> pixel p = lane&15, filter channel fc = lane>>4 (F = 2)
    const int p  = m;
    const int fc = lane >> 4;
    const float gi = sw[p * 16 + 0 + fc];
    const float gf = sw[p * 16 + 2 + fc];
    const float gc = sw[p * 16 + 4 + fc];
    const float go = sw[p * 16 + 6 + fc];

    const int cidx = ((b * Fsz + fc) * Hsz + y) * Wsz + x0 + p;
    const float cprev = (t > 0) ? cws[cidx] : 0.0f;

    const float i_ = hsig(gi);
    const float f_ = hsig(gf);
    const float cn = f_ * cprev + i_ * tanhf(gc);
    const float o_ = hsig(go);
    const float hn = o_ * tanhf(cn);

    cws[cidx] = cn;
    out[(size_t)((b * Tsz + t) * Fsz + fc) * HW + y * Wsz + x0 + p] = hn;
}

extern "C" void kernel_launch(void* const* d_in, const int* in_sizes, int n_in,
                              void* d_out, int out_size, void* d_ws, size_t ws_size,
                              hipStream_t stream) {
    const float* x  = (const float*)d_in[0];
    const float* Wx = (const float*)d_in[1];
    const float* Wh = (const float*)d_in[2];
    float* out = (float*)d_out;
    float* cws = (float*)d_ws;   // needs B*F*H*W*4 = 1 MB

    dim3 grid(1024);   // 8192 tiles / 8 waves per block
    dim3 block(256);
    for (int t = 0; t < Tsz; ++t) {
        convlstm_step<<<grid, block, 0, stream>>>(x, Wx, Wh, out, cws, t);
    }
}